// Rescale_19610820673536
// MI455X (gfx1250) — compile-verified
//
#include <hip/hip_runtime.h>

// out[i, j] = x[i, j] * zzw[j]^2   (x: [B,10] fp32, row-major; out same shape)
//
// Memory-bound streaming kernel: 320 MB of traffic, ~13.7 us floor at 23.3 TB/s.
// Strategy: b128 coalesced loads/stores with non-temporal hints (stream-once
// data must not thrash L2), per-thread scale float4 fetched once from a tiny
// LDS table, grid-stride divisible by 5 so the column phase is loop-invariant.

typedef float float4v __attribute__((ext_vector_type(4)));

#define BLOCK 256          // 8 wave32s per workgroup
#define GRID  2500         // BLOCK*GRID = 640,000 float4s per sweep, % 5 == 0

__global__ __launch_bounds__(BLOCK) void rescale_scale10_kernel(
    const float* __restrict__ x,
    const float* __restrict__ zzw,
    float* __restrict__ out,
    int n4,                 // number of full float4 vectors
    int n)                  // total element count (for tail)
{
    // lscale[t] = zzw[t % 10]^2 for t in [0,20): five aligned float4 scale
    // vectors covering the (4q+k) mod 10 column pattern, period 5 in q.
    __shared__ __align__(16) float lscale[20];

    const int t = threadIdx.x;
    if (t < 20) {
        float z = zzw[t % 10];
        lscale[t] = z * z;
    }
    __syncthreads();

    const float4v* __restrict__ x4 = (const float4v*)x;
    float4v* __restrict__ o4       = (float4v*)out;

    const int q0 = blockIdx.x * BLOCK + t;
    // stride % 5 == 0  ->  q mod 5 is invariant across the grid-stride loop
    const int r  = q0 % 5;
    const float4v s = *(const float4v*)&lscale[r * 4];

    const int stride = BLOCK * GRID;
    #pragma unroll 4
    for (int q = q0; q < n4; q += stride) {
        float4v v = __builtin_nontemporal_load(&x4[q]);   // global_load_b128, NT
        v = v * s;
        __builtin_nontemporal_store(v, &o4[q]);           // global_store_b128, NT
    }

    // Scalar tail (n not divisible by 4). 40M is exact, but stay generic.
    const int tail_base = n4 * 4;
    const int rem = n - tail_base;
    if (blockIdx.x == 0 && t < rem) {
        const int i = tail_base + t;
        out[i] = x[i] * lscale[i % 10];
    }
}

extern "C" void kernel_launch(void* const* d_in, const int* in_sizes, int n_in,
                              void* d_out, int out_size, void* d_ws, size_t ws_size,
                              hipStream_t stream) {
    const float* x   = (const float*)d_in[0];   // [B*10] fp32
    const float* zzw = (const float*)d_in[1];   // [10]   fp32
    float* out       = (float*)d_out;           // [B*10] fp32

    const int n  = out_size;                    // 40,000,000
    const int n4 = n / 4;

    rescale_scale10_kernel<<<GRID, BLOCK, 0, stream>>>(x, zzw, out, n4, n);
}